// LDDF_26233660244336
// MI455X (gfx1250) — compile-verified
//
#include <hip/hip_runtime.h>

// ---------------------------------------------------------------------------
// Fused time/freq attention block (channel attn -> spatial attn -> 1x1 fuse ->
// BatchNorm(batch stats) -> ReLU -> residual) for B=131072, C=2, L=128 fp32.
//
// Memory-bound: ~640 MB total traffic with the recompute scheme -> ~28 us at
// 23.3 TB/s.  WMMA is inapplicable (channel dims 2/4 << 16x16 tile); the
// CDNA5-specific feature used is the async global->LDS copy engine
// (GLOBAL_LOAD_ASYNC_TO_LDS_B128 + s_wait_asynccnt), double-buffered per wave.
// ---------------------------------------------------------------------------

#define B_ROWS   131072
#define L_LEN    128
#define ROW_STR  256          // 2 channels * 128 floats per row per tensor
#define LANES    32
#define WPB      8            // waves per block
#define BLOCK    256
#define NBLK     4096         // blocks in main kernels
#define RPW      4            // rows per wave: NBLK*WPB*RPW == B_ROWS
#define EPSBN    1e-5f

#ifndef __has_builtin
#define __has_builtin(x) 0
#endif

#if defined(__HIP_DEVICE_COMPILE__) && defined(__gfx1250__) && \
    __has_builtin(__builtin_amdgcn_global_load_async_to_lds_b128)
#define USE_ASYNC 1
// Exact pointee type per hipcc diagnostic: int __vector(4) in AS1 / AS3.
typedef int v4i __attribute__((vector_size(4 * sizeof(int))));
typedef __attribute__((address_space(1))) v4i GV4;   // global int4
typedef __attribute__((address_space(3))) v4i LV4;   // LDS int4
#else
#define USE_ASYNC 0
#endif

struct __align__(16) V4 { float v[4]; };

__device__ __forceinline__ float wred32(float x) {
#pragma unroll
  for (int m = 16; m >= 1; m >>= 1) x += __shfl_xor(x, m, 32);
  return x;
}
__device__ __forceinline__ float sigm(float x) { return 1.0f / (1.0f + __expf(-x)); }
__device__ __forceinline__ float shup1(float x, int lane) {
  float r = __shfl_up(x, 1, 32); return lane == 0 ? 0.0f : r;
}
__device__ __forceinline__ float shdn1(float x, int lane) {
  float r = __shfl_down(x, 1, 32); return lane == 31 ? 0.0f : r;
}

// Per-row chain up to (pre-BN) fused outputs. Lane owns L positions
// lane*4 .. lane*4+3.  Channels: 0,1 = time; 2,3 = freq.
__device__ __forceinline__ void row_core(
    const V4& t0, const V4& t1, const V4& f0, const V4& f1,
    const float* ca1, const float* ca2, const float* ka, const float* km,
    const float* wf, int lane, float fu0[4], float fu1[4])
{
  // ---- channel attention: mean over L per channel -> 4->2 relu -> 2->4 sigm
  const float inv_l = 1.0f / (float)L_LEN;
  float p0 = wred32((t0.v[0] + t0.v[1]) + (t0.v[2] + t0.v[3])) * inv_l;
  float p1 = wred32((t1.v[0] + t1.v[1]) + (t1.v[2] + t1.v[3])) * inv_l;
  float p2 = wred32((f0.v[0] + f0.v[1]) + (f0.v[2] + f0.v[3])) * inv_l;
  float p3 = wred32((f1.v[0] + f1.v[1]) + (f1.v[2] + f1.v[3])) * inv_l;

  float h0 = fmaxf(0.0f, p0*ca1[0] + p1*ca1[1] + p2*ca1[2] + p3*ca1[3]);
  float h1 = fmaxf(0.0f, p0*ca1[4] + p1*ca1[5] + p2*ca1[6] + p3*ca1[7]);
  float a0 = sigm(h0*ca2[0] + h1*ca2[1]);
  float a1 = sigm(h0*ca2[2] + h1*ca2[3]);
  float a2 = sigm(h0*ca2[4] + h1*ca2[5]);
  float a3 = sigm(h0*ca2[6] + h1*ca2[7]);

  float x0[4], x1[4], x2[4], x3[4], avg[4], mx[4];
#pragma unroll
  for (int j = 0; j < 4; ++j) {
    x0[j] = t0.v[j] * a0;  x1[j] = t1.v[j] * a1;
    x2[j] = f0.v[j] * a2;  x3[j] = f1.v[j] * a3;
    avg[j] = (x0[j] + x1[j] + x2[j] + x3[j]) * 0.25f;
    mx[j]  = fmaxf(fmaxf(x0[j], x1[j]), fmaxf(x2[j], x3[j]));
  }

  // ---- spatial attention: 7-tap conv (pad 3) over L; halo via lane shuffles
  float wa[10], wm[10];
  wa[0] = shup1(avg[1], lane); wa[1] = shup1(avg[2], lane); wa[2] = shup1(avg[3], lane);
  wa[3] = avg[0]; wa[4] = avg[1]; wa[5] = avg[2]; wa[6] = avg[3];
  wa[7] = shdn1(avg[0], lane); wa[8] = shdn1(avg[1], lane); wa[9] = shdn1(avg[2], lane);
  wm[0] = shup1(mx[1], lane);  wm[1] = shup1(mx[2], lane);  wm[2] = shup1(mx[3], lane);
  wm[3] = mx[0];  wm[4] = mx[1];  wm[5] = mx[2];  wm[6] = mx[3];
  wm[7] = shdn1(mx[0], lane);  wm[8] = shdn1(mx[1], lane);  wm[9] = shdn1(mx[2], lane);

#pragma unroll
  for (int j = 0; j < 4; ++j) {
    float c = 0.0f;
#pragma unroll
    for (int t = 0; t < 7; ++t)
      c = fmaf(wa[j + t], ka[t], fmaf(wm[j + t], km[t], c));
    float s = sigm(c);
    // ---- 1x1 fusion conv (pre-BN)
    fu0[j] = s * (x0[j]*wf[0] + x1[j]*wf[1] + x2[j]*wf[2] + x3[j]*wf[3]);
    fu1[j] = s * (x0[j]*wf[4] + x1[j]*wf[5] + x2[j]*wf[6] + x3[j]*wf[7]);
  }
}

// STATS=1: accumulate per-block {sum,sumsq} of fused per output channel.
// STATS=0: apply BN scale/shift (from ws), ReLU, residual; write out.
template <int STATS>
__global__ __launch_bounds__(BLOCK) void lddf_main(
    const float* __restrict__ tf, const float* __restrict__ ff,
    const float* __restrict__ w_ca1, const float* __restrict__ w_ca2,
    const float* __restrict__ w_sa, const float* __restrict__ w_fuse,
    const float* __restrict__ rw_p, float* __restrict__ out,
    float* __restrict__ ws)
{
  const int lane = threadIdx.x & (LANES - 1);
  const int wv   = threadIdx.x >> 5;

  float ca1[8], ca2[8], wf[8], ka[7], km[7];
#pragma unroll
  for (int i = 0; i < 8; ++i) { ca1[i] = w_ca1[i]; ca2[i] = w_ca2[i]; wf[i] = w_fuse[i]; }
#pragma unroll
  for (int i = 0; i < 7; ++i) { ka[i] = w_sa[i]; km[i] = w_sa[7 + i]; }

  float sc0 = 0, sh0 = 0, sc1 = 0, sh1 = 0, rw = 0;
  if constexpr (!STATS) {
    const float* p = ws + 4 * NBLK;
    sc0 = p[0]; sh0 = p[1]; sc1 = p[2]; sh1 = p[3]; rw = rw_p[0];
  }

  float acc0 = 0, acc1 = 0, acc2 = 0, acc3 = 0;
  const int row0 = (blockIdx.x * WPB + wv) * RPW;

#if USE_ASYNC
  __shared__ float abuf[WPB][2][512];   // per-wave double buffer: 4 x 128 floats
  float* buf = &abuf[wv][0][0];
  auto issue = [&](int row, float* dst) {
    const float* gt = tf + (size_t)row * ROW_STR + lane * 4;
    const float* gf = ff + (size_t)row * ROW_STR + lane * 4;
    __builtin_amdgcn_global_load_async_to_lds_b128(
        (GV4*)const_cast<float*>(gt),       (LV4*)(dst + lane * 4),       0, 0);
    __builtin_amdgcn_global_load_async_to_lds_b128(
        (GV4*)const_cast<float*>(gt + 128), (LV4*)(dst + 128 + lane * 4), 0, 0);
    __builtin_amdgcn_global_load_async_to_lds_b128(
        (GV4*)const_cast<float*>(gf),       (LV4*)(dst + 256 + lane * 4), 0, 0);
    __builtin_amdgcn_global_load_async_to_lds_b128(
        (GV4*)const_cast<float*>(gf + 128), (LV4*)(dst + 384 + lane * 4), 0, 0);
  };
  issue(row0, buf);
  int pb = 0;
#endif

#pragma unroll
  for (int i = 0; i < RPW; ++i) {
    const int row = row0 + i;
    V4 t0, t1, f0, f1;
#if USE_ASYNC
    if (i + 1 < RPW) {
      issue(row + 1, buf + (pb ^ 1) * 512);            // prefetch into other buffer
      asm volatile("s_wait_asynccnt 4" ::: "memory");  // oldest 4 (buf[pb]) done
    } else {
      asm volatile("s_wait_asynccnt 0" ::: "memory");
    }
    const float* bsrc = buf + pb * 512;
    t0 = *(const V4*)(bsrc + lane * 4);
    t1 = *(const V4*)(bsrc + 128 + lane * 4);
    f0 = *(const V4*)(bsrc + 256 + lane * 4);
    f1 = *(const V4*)(bsrc + 384 + lane * 4);
    pb ^= 1;
#else
    const float* gt = tf + (size_t)row * ROW_STR + lane * 4;
    const float* gf = ff + (size_t)row * ROW_STR + lane * 4;
    t0 = *(const V4*)gt;        t1 = *(const V4*)(gt + 128);
    f0 = *(const V4*)gf;        f1 = *(const V4*)(gf + 128);
#endif

    float fu0[4], fu1[4];
    row_core(t0, t1, f0, f1, ca1, ca2, ka, km, wf, lane, fu0, fu1);

    if constexpr (STATS) {
#pragma unroll
      for (int j = 0; j < 4; ++j) {
        acc0 += fu0[j]; acc1 += fu0[j] * fu0[j];
        acc2 += fu1[j]; acc3 += fu1[j] * fu1[j];
      }
    } else {
      V4 o0, o1;
#pragma unroll
      for (int j = 0; j < 4; ++j) {
        o0.v[j] = fmaxf(fu0[j] * sc0 + sh0, 0.0f) + rw * t0.v[j];
        o1.v[j] = fmaxf(fu1[j] * sc1 + sh1, 0.0f) + rw * t1.v[j];
      }
      float* op = out + (size_t)row * ROW_STR + lane * 4;
      *(V4*)op = o0;
      *(V4*)(op + 128) = o1;
    }
  }

  if constexpr (STATS) {
    acc0 = wred32(acc0); acc1 = wred32(acc1);
    acc2 = wred32(acc2); acc3 = wred32(acc3);
    __shared__ float sred[4][WPB];
    if (lane == 0) { sred[0][wv] = acc0; sred[1][wv] = acc1;
                     sred[2][wv] = acc2; sred[3][wv] = acc3; }
    __syncthreads();
    if (wv == 0) {
      float b0 = lane < WPB ? sred[0][lane] : 0.0f;
      float b1 = lane < WPB ? sred[1][lane] : 0.0f;
      float b2 = lane < WPB ? sred[2][lane] : 0.0f;
      float b3 = lane < WPB ? sred[3][lane] : 0.0f;
      b0 = wred32(b0); b1 = wred32(b1); b2 = wred32(b2); b3 = wred32(b3);
      if (lane == 0) {
        ws[blockIdx.x]            = b0;   // sum ch0
        ws[NBLK + blockIdx.x]     = b1;   // sumsq ch0
        ws[2 * NBLK + blockIdx.x] = b2;   // sum ch1
        ws[3 * NBLK + blockIdx.x] = b3;   // sumsq ch1
      }
    }
  }
}

__global__ __launch_bounds__(256) void lddf_reduce(
    float* __restrict__ ws, const float* __restrict__ gamma,
    const float* __restrict__ beta)
{
  const int t = threadIdx.x, lane = t & 31, wv = t >> 5;
  float a0 = 0, a1 = 0, a2 = 0, a3 = 0;
  for (int g = t; g < NBLK; g += 256) {
    a0 += ws[g];            a1 += ws[NBLK + g];
    a2 += ws[2 * NBLK + g]; a3 += ws[3 * NBLK + g];
  }
  a0 = wred32(a0); a1 = wred32(a1); a2 = wred32(a2); a3 = wred32(a3);
  __shared__ float sm[4][8];
  if (lane == 0) { sm[0][wv] = a0; sm[1][wv] = a1; sm[2][wv] = a2; sm[3][wv] = a3; }
  __syncthreads();
  if (t == 0) {
    float S0 = 0, Q0 = 0, S1 = 0, Q1 = 0;
    for (int i = 0; i < 8; ++i) { S0 += sm[0][i]; Q0 += sm[1][i];
                                  S1 += sm[2][i]; Q1 += sm[3][i]; }
    const float invN = 1.0f / (float)((long long)B_ROWS * L_LEN);
    float m0 = S0 * invN, m1 = S1 * invN;
    float v0 = fmaxf(Q0 * invN - m0 * m0, 0.0f);
    float v1 = fmaxf(Q1 * invN - m1 * m1, 0.0f);
    float sc0 = gamma[0] * rsqrtf(v0 + EPSBN);
    float sc1 = gamma[1] * rsqrtf(v1 + EPSBN);
    float* o = ws + 4 * NBLK;
    o[0] = sc0; o[1] = beta[0] - m0 * sc0;
    o[2] = sc1; o[3] = beta[1] - m1 * sc1;
  }
}

extern "C" void kernel_launch(void* const* d_in, const int* in_sizes, int n_in,
                              void* d_out, int out_size, void* d_ws, size_t ws_size,
                              hipStream_t stream) {
  const float* tf     = (const float*)d_in[0];
  const float* ff     = (const float*)d_in[1];
  const float* w_ca1  = (const float*)d_in[2];
  const float* w_ca2  = (const float*)d_in[3];
  const float* w_sa   = (const float*)d_in[4];
  const float* w_fuse = (const float*)d_in[5];
  const float* gamma  = (const float*)d_in[6];
  const float* beta   = (const float*)d_in[7];
  const float* rw     = (const float*)d_in[8];
  float* out = (float*)d_out;
  float* ws  = (float*)d_ws;   // needs (4*NBLK + 4) * 4 bytes = 64 KB + 16 B

  lddf_main<1><<<NBLK, BLOCK, 0, stream>>>(tf, ff, w_ca1, w_ca2, w_sa, w_fuse,
                                           rw, out, ws);
  lddf_reduce<<<1, 256, 0, stream>>>(ws, gamma, beta);
  lddf_main<0><<<NBLK, BLOCK, 0, stream>>>(tf, ff, w_ca1, w_ca2, w_sa, w_fuse,
                                           rw, out, ws);
}